// RNN_19275813225189
// MI455X (gfx1250) — compile-verified
//
#include <hip/hip_runtime.h>

typedef __attribute__((ext_vector_type(2))) float v2f;
typedef __attribute__((ext_vector_type(8))) float v8f;

#define KC   32   // K-chunk staged in LDS
#define LDSW 36   // padded LDS row stride (floats): conflict-free fragment reads, 16B-aligned stores

// Tile GEMM:  C = op(A * W^T + bias [+ C])
//   A: M x K, row stride lda (fp32, row-major)
//   W: N x K, row-major (weight matrices are (out,in), so C[m,n] = sum_k A[m,k]*W[n,k])
//   C: M x N, row stride ldc
// Block tile: (NWAVES*16) rows x 64 cols. Each wave: 16 rows x 64 cols via 4 f32 WMMA accumulators.
// MODE 0: C = A*W^T + bias
// MODE 1: C = relu(C + A*W^T + bias)   (C holds precomputed xw_t for the RNN recurrence)
template <int NWAVES, int MODE>
__global__ __launch_bounds__(NWAVES * 32)
void wmma_gemm_kernel(const float* __restrict__ A, long lda,
                      const float* __restrict__ W,
                      const float* __restrict__ bias,
                      float* __restrict__ C, long ldc,
                      int K)
{
    __shared__ float Ws[64 * LDSW];

    const int tid  = threadIdx.x;
    const int wave = tid >> 5;
    const int lane = tid & 31;
    const int r    = lane & 15;   // M-row (A frag) / N-col (B,C frags) within tile
    const int hi   = lane >> 4;   // selects K pair {0,1} vs {2,3} for A/B frags

    const int nblk = blockIdx.x * 64;
    const int m0   = blockIdx.y * (NWAVES * 16) + wave * 16;

    const float* Arow = A + (long)(m0 + r) * lda;

    v8f acc[4] = {};

    for (int k0 = 0; k0 < K; k0 += KC) {
        __syncthreads();
        // Cooperative stage of W[nblk..nblk+63][k0..k0+KC) into LDS (float4 loads/stores).
        for (int idx = tid * 4; idx < 64 * KC; idx += NWAVES * 32 * 4) {
            const int n  = idx >> 5;      // KC == 32
            const int kk = idx & 31;
            const float* p = W + (long)(nblk + n) * K + k0 + kk;
            float4 v = *(const float4*)p;
            *(float4*)&Ws[n * LDSW + kk] = v;
            if (k0 + KC < K) __builtin_prefetch(p + KC, 0, 3);  // global_prefetch_b8 next chunk
        }
        __syncthreads();

#pragma unroll
        for (int kk = 0; kk < KC; kk += 4) {
            // A fragment 16x4 f32: lane r holds row m0+r, K = k0+kk+2*hi .. +1 (contiguous float2)
            v2f a = *(const v2f*)(Arow + k0 + kk + 2 * hi);
#pragma unroll
            for (int s = 0; s < 4; ++s) {
                // B fragment 4x16 (K x N): lane r holds col nblk+s*16+r, K pair per `hi`
                const float* bp = &Ws[(s * 16 + r) * LDSW + kk + 2 * hi];
                v2f b;
                b.x = bp[0];
                b.y = bp[1];
                acc[s] = __builtin_amdgcn_wmma_f32_16x16x4_f32(
                    false, a, false, b, (short)0, acc[s], false, false);
            }
        }
    }
    __syncthreads();

    // Epilogue: C/D layout — VGPR j: lanes 0-15 row m0+j, lanes 16-31 row m0+8+j; col = n + lane%16
#pragma unroll
    for (int s = 0; s < 4; ++s) {
        const int col = nblk + s * 16 + r;
        const float bv = bias[col];
#pragma unroll
        for (int j = 0; j < 8; ++j) {
            const int row = m0 + hi * 8 + j;
            float* cp = C + (long)row * ldc + col;
            float v = acc[s][j] + bv;
            if (MODE == 1) {
                v += *cp;                  // += xw_t (stored in place)
                v = v > 0.f ? v : 0.f;     // relu
            }
            *cp = v;
        }
    }
}

// t = 0 step: h_0 = relu(xw_0 + b_h), in place on hidden[:,0,:] (row stride T*H).
__global__ void rnn_t0_relu_kernel(float* __restrict__ hid, const float* __restrict__ b_h)
{
    const int idx = blockIdx.x * blockDim.x + threadIdx.x;   // 0 .. B*H-1
    const int b = idx >> 10;                                 // H = 1024
    const int h = idx & 1023;
    float* p = hid + (long)b * (512L * 1024L) + h;
    float v = *p + b_h[h];
    *p = v > 0.f ? v : 0.f;
}

extern "C" void kernel_launch(void* const* d_in, const int* in_sizes, int n_in,
                              void* d_out, int out_size, void* d_ws, size_t ws_size,
                              hipStream_t stream)
{
    (void)in_sizes; (void)n_in; (void)out_size; (void)d_ws; (void)ws_size;

    const float* x    = (const float*)d_in[0];
    const float* W_in = (const float*)d_in[1];
    const float* b_in = (const float*)d_in[2];
    const float* W_h  = (const float*)d_in[3];
    const float* b_h  = (const float*)d_in[4];
    const float* W_o  = (const float*)d_in[5];
    const float* b_o  = (const float*)d_in[6];

    const int B = 256, T = 512, I = 512, H = 1024, O = 128;

    float* out = (float*)d_out;                       // (B,T,O)
    float* hid = out + (size_t)B * T * O;             // (B,T,H)

    // Phase 1: xw = x @ W_in^T + b_in  -> hidden region (in place).  M=B*T, N=H, K=I.
    {
        dim3 grid(H / 64, (B * T) / (8 * 16));
        wmma_gemm_kernel<8, 0><<<grid, 8 * 32, 0, stream>>>(x, (long)I, W_in, b_in,
                                                            hid, (long)H, I);
    }

    // Phase 2a: t = 0, h_0 = relu(xw_0 + b_h)
    rnn_t0_relu_kernel<<<(B * H) / 256, 256, 0, stream>>>(hid, b_h);

    // Phase 2b: recurrence, t = 1..T-1:  hid[:,t,:] = relu(hid[:,t,:] + hid[:,t-1,:] @ W_h^T + b_h)
    // A rows are strided by T*H (batch-major hidden layout). M=B, N=H, K=H.
    for (int t = 1; t < T; ++t) {
        dim3 grid(H / 64, B / (2 * 16));
        wmma_gemm_kernel<2, 1><<<grid, 2 * 32, 0, stream>>>(
            hid + (size_t)(t - 1) * H, (long)T * H, W_h, b_h,
            hid + (size_t)t * H,       (long)T * H, H);
    }

    // Phase 3: output = hidden @ W_o^T + b_o.  M=B*T, N=O, K=H.
    {
        dim3 grid(O / 64, (B * T) / (8 * 16));
        wmma_gemm_kernel<8, 0><<<grid, 8 * 32, 0, stream>>>(hid, (long)H, W_o, b_o,
                                                            out, (long)O, H);
    }
}